// GCNModel_21388937134273
// MI455X (gfx1250) — compile-verified
//
#include <hip/hip_runtime.h>
#include <stdint.h>

#define TPB 256

// Use inline CDNA5 asm for the async global->LDS staging path (bypasses the
// clang builtin whose prototype differs across toolchains).
#if defined(__gfx1250__)
# define USE_ASYNC_LDS 1
#else
# define USE_ASYNC_LDS 0
#endif

// ---------------- utility kernels ----------------

__global__ __launch_bounds__(TPB) void k_fill0(float* __restrict__ p, int n) {
  int i = blockIdx.x * TPB + threadIdx.x;
  if (i < n) p[i] = 0.0f;
}

__global__ __launch_bounds__(TPB) void k_count(const int* __restrict__ idx, int E,
                                               float* __restrict__ deg) {
  int i = blockIdx.x * TPB + threadIdx.x;
  if (i < E) atomicAdd(&deg[idx[i]], 1.0f);
}

// deg -> clip(deg,1)^-0.5
__global__ __launch_bounds__(TPB) void k_finalize_deg(float* __restrict__ deg, int n) {
  int i = blockIdx.x * TPB + threadIdx.x;
  if (i < n) deg[i] = rsqrtf(fmaxf(deg[i], 1.0f));
}

// emb = out = concat(user, item)   (float4 granularity)
__global__ __launch_bounds__(TPB) void k_init_concat(const float4* __restrict__ u,
                                                     const float4* __restrict__ it,
                                                     int nu4, int ntot4,
                                                     float4* __restrict__ emb,
                                                     float4* __restrict__ out) {
  int i = blockIdx.x * TPB + threadIdx.x;
  if (i < ntot4) {
    float4 v = (i < nu4) ? u[i] : it[i - nu4];
    emb[i] = v;
    out[i] = v;
  }
}

// emb *= deg_out^-0.5 (per row), agg = 0   (float4 granularity, 16 per row)
__global__ __launch_bounds__(TPB) void k_scale_zero(float4* __restrict__ emb,
                                                    const float* __restrict__ dinv,
                                                    float4* __restrict__ agg, int n4) {
  int i = blockIdx.x * TPB + threadIdx.x;
  if (i < n4) {
    float s = dinv[i >> 4];
    float4 v = emb[i];
    v.x *= s; v.y *= s; v.z *= s; v.w *= s;
    emb[i] = v;
    agg[i] = make_float4(0.f, 0.f, 0.f, 0.f);
  }
}

// scatter-add SpMM: 16 threads per edge, each moves one float4 of the 256B row.
// Source row is one L2 cache line; whole feature matrix (76.8MB) is L2-resident.
__global__ __launch_bounds__(TPB) void k_edge(const int* __restrict__ src,
                                              const int* __restrict__ dst, int E,
                                              const float* __restrict__ x,
                                              float* __restrict__ agg) {
  int i = blockIdx.x * TPB + threadIdx.x;
  int tot = E * 16;
  if (i < tot) {
    int e = i >> 4;
    int t = (i & 15) * 4;
    int s = src[e];
    int d = dst[e];
    float4 v = *reinterpret_cast<const float4*>(x + (size_t)s * 64 + t);
    float* a = agg + (size_t)d * 64 + t;
    atomicAdd(a + 0, v.x);   // no-return global_atomic_add_f32
    atomicAdd(a + 1, v.y);
    atomicAdd(a + 2, v.z);
    atomicAdd(a + 3, v.w);
  }
}

// Per-row epilogue: rst = agg * deg_in^-0.5 ; LeakyReLU(0.5) ; emb = rst ;
// sum += rst / max(||rst||2, 1e-12).  One wave32 per 64-float row, 8 rows/block.
// Rows are staged into LDS with CDNA5 async global->LDS copies (inline asm).
__global__ __launch_bounds__(TPB) void k_row_post(const float* __restrict__ agg,
                                                  const float* __restrict__ dinv,
                                                  float* __restrict__ emb,
                                                  float* __restrict__ sum_out,
                                                  int nrows) {
  __shared__ float tile[8 * 64];
  int row0 = blockIdx.x * 8;
  int tid = threadIdx.x;
  int nrow_blk = nrows - row0;
  if (nrow_blk > 8) nrow_blk = 8;
  int nf4 = nrow_blk * 16;  // float4 chunks to stage

  if (tid < nf4) {
    const float* g = agg + (size_t)row0 * 64 + tid * 4;
#if USE_ASYNC_LDS
    // VDST = per-lane LDS byte address (low 32 bits of flat shared pointer),
    // VADDR = per-lane 64-bit global address.
    unsigned int lds_addr = (unsigned int)(uintptr_t)(&tile[tid * 4]);
    unsigned long long gaddr = (unsigned long long)(uintptr_t)g;
    asm volatile("global_load_async_to_lds_b128 %0, %1, off"
                 :
                 : "v"(lds_addr), "v"(gaddr)
                 : "memory");
#else
    *reinterpret_cast<float4*>(&tile[tid * 4]) = *reinterpret_cast<const float4*>(g);
#endif
  }
#if USE_ASYNC_LDS
  asm volatile("s_wait_asynccnt 0" ::: "memory");
#endif
  __syncthreads();

  int wave = tid >> 5;
  int lane = tid & 31;
  int r = row0 + wave;
  if (r < nrows) {
    float di = dinv[r];
    float2 t2 = *reinterpret_cast<const float2*>(&tile[wave * 64 + lane * 2]);
    float a0 = t2.x * di;
    float a1 = t2.y * di;
    a0 = (a0 > 0.f) ? a0 : 0.5f * a0;  // LeakyReLU(0.5)
    a1 = (a1 > 0.f) ? a1 : 0.5f * a1;
    float ss = a0 * a0 + a1 * a1;
    // wave32 butterfly reduction
    #pragma unroll
    for (int off = 16; off >= 1; off >>= 1) ss += __shfl_xor(ss, off, 32);
    float inv = 1.0f / fmaxf(sqrtf(ss), 1e-12f);
    size_t base = (size_t)r * 64 + lane * 2;
    emb[base + 0] = a0;
    emb[base + 1] = a1;
    sum_out[base + 0] += a0 * inv;
    sum_out[base + 1] += a1 * inv;
  }
}

// ---------------- host orchestration ----------------

static inline int nblk(long n) { return (int)((n + TPB - 1) / TPB); }

extern "C" void kernel_launch(void* const* d_in, const int* in_sizes, int n_in,
                              void* d_out, int out_size, void* d_ws, size_t ws_size,
                              hipStream_t stream) {
  const float* user_emb = (const float*)d_in[0];
  const float* item_emb = (const float*)d_in[1];
  const int* ui_src = (const int*)d_in[2];
  const int* ui_dst = (const int*)d_in[3];
  const int* uu_src = (const int*)d_in[4];
  const int* uu_dst = (const int*)d_in[5];

  const int D = 64;
  const int NU = in_sizes[0] / D;      // 100000
  const int NI = in_sizes[1] / D;      // 200000
  const int NA = NU + NI;              // 300000
  const int EUI = in_sizes[2];         // 3.2M
  const int EUU = in_sizes[4];         // 1.6M
  const int N_LAYERS = 3, S_LAYERS = 2;

  float* out_ui = (float*)d_out;                    // [NA, 64]
  float* out_uu = out_ui + (size_t)NA * D;          // [NU, 64]

  // workspace carve-out (256B aligned slices)
  char* w = (char*)d_ws;
  auto take = [&](size_t bytes) -> float* {
    float* p = (float*)w;
    w += (bytes + 255) & ~(size_t)255;
    return p;
  };
  float* emb = take((size_t)NA * D * sizeof(float));   // 76.8 MB
  float* agg = take((size_t)NA * D * sizeof(float));   // 76.8 MB
  float* dAo = take((size_t)NA * sizeof(float));
  float* dAi = take((size_t)NA * sizeof(float));
  float* dUo = take((size_t)NU * sizeof(float));
  float* dUi = take((size_t)NU * sizeof(float));

  // --- degrees (constant across layers: compute inv-sqrt once) ---
  k_fill0<<<nblk(NA), TPB, 0, stream>>>(dAo, NA);
  k_fill0<<<nblk(NA), TPB, 0, stream>>>(dAi, NA);
  k_fill0<<<nblk(NU), TPB, 0, stream>>>(dUo, NU);
  k_fill0<<<nblk(NU), TPB, 0, stream>>>(dUi, NU);
  k_count<<<nblk(EUI), TPB, 0, stream>>>(ui_src, EUI, dAo);
  k_count<<<nblk(EUI), TPB, 0, stream>>>(ui_dst, EUI, dAi);
  k_count<<<nblk(EUU), TPB, 0, stream>>>(uu_src, EUU, dUo);
  k_count<<<nblk(EUU), TPB, 0, stream>>>(uu_dst, EUU, dUi);
  k_finalize_deg<<<nblk(NA), TPB, 0, stream>>>(dAo, NA);
  k_finalize_deg<<<nblk(NA), TPB, 0, stream>>>(dAi, NA);
  k_finalize_deg<<<nblk(NU), TPB, 0, stream>>>(dUo, NU);
  k_finalize_deg<<<nblk(NU), TPB, 0, stream>>>(dUi, NU);

  // --- user-item propagation over all nodes ---
  {
    int nu4 = NU * (D / 4), ntot4 = NA * (D / 4);
    k_init_concat<<<nblk(ntot4), TPB, 0, stream>>>(
        (const float4*)user_emb, (const float4*)item_emb, nu4, ntot4,
        (float4*)emb, (float4*)out_ui);
    for (int l = 0; l < N_LAYERS; ++l) {
      k_scale_zero<<<nblk(ntot4), TPB, 0, stream>>>((float4*)emb, dAo, (float4*)agg, ntot4);
      k_edge<<<nblk((long)EUI * 16), TPB, 0, stream>>>(ui_src, ui_dst, EUI, emb, agg);
      k_row_post<<<(NA + 7) / 8, TPB, 0, stream>>>(agg, dAi, emb, out_ui, NA);
    }
  }

  // --- user-user propagation ---
  {
    int nu4 = NU * (D / 4);
    // emb = out_uu = user_emb (reuse concat kernel with item count 0)
    k_init_concat<<<nblk(nu4), TPB, 0, stream>>>(
        (const float4*)user_emb, (const float4*)user_emb, nu4, nu4,
        (float4*)emb, (float4*)out_uu);
    for (int l = 0; l < S_LAYERS; ++l) {
      k_scale_zero<<<nblk(nu4), TPB, 0, stream>>>((float4*)emb, dUo, (float4*)agg, nu4);
      k_edge<<<nblk((long)EUU * 16), TPB, 0, stream>>>(uu_src, uu_dst, EUU, emb, agg);
      k_row_post<<<(NU + 7) / 8, TPB, 0, stream>>>(agg, dUi, emb, out_uu, NU);
    }
  }
}